// DecouplingSpecificSpecificLoss_81252191305955
// MI455X (gfx1250) — compile-verified
//
#include <hip/hip_runtime.h>

// ---------------------------------------------------------------------------
// DecouplingSpecificSpecificLoss on gfx1250 (MI455X)
//   X: [16384, 2048] f32.  4096 chunks of NUM_VIEWS=4 rows.
//   Per wave: 16 rows (4 chunks) -> 16x16 Gram via V_WMMA_F32_16X16X4_F32,
//   chained over K=2048 (512 WMMAs). Epilogue normalizes by diagonal,
//   clips, -log(1-sim), mean(1/16), deterministic 2-pass reduction.
//   Memory-bound: 128 MiB @ 23.3 TB/s ~= 5.8 us floor; WMMA math is free.
// ---------------------------------------------------------------------------

typedef __attribute__((ext_vector_type(2))) float v2f;
typedef __attribute__((ext_vector_type(8))) float v8f;

#define DIMD            2048
#define ROWS_PER_WAVE   16
#define WAVES_PER_BLOCK 8
#define BLOCK_THREADS   (WAVES_PER_BLOCK * 32)

#define CLAMP_MIN 0.0005f
#define CLAMP_MAX 0.9995f
#define NORM_EPS  1e-12f

__device__ __forceinline__ v8f wmma_f32(v2f a, v2f b, v8f c) {
  // 8-arg form: (neg_a, A, neg_b, B, c_mod, C, reuse_a, reuse_b)
  return __builtin_amdgcn_wmma_f32_16x16x4_f32(false, a, false, b,
                                               (short)0, c, false, false);
}

__global__ __launch_bounds__(BLOCK_THREADS)
void gram_loss_kernel(const float* __restrict__ X, float* __restrict__ partial) {
  __shared__ float diag_s[WAVES_PER_BLOCK][16];
  __shared__ float invn_s[WAVES_PER_BLOCK][16];

  const int lane  = threadIdx.x & 31;
  const int wid   = threadIdx.x >> 5;
  const int gwave = blockIdx.x * WAVES_PER_BLOCK + wid;

  // A-matrix (and identical B-matrix) f32 16x4 layout:
  //   lanes 0-15 : row = lane,    K offsets {0,1}
  //   lanes16-31 : row = lane-16, K offsets {2,3}
  const int  row  = gwave * ROWS_PER_WAVE + (lane & 15);
  const int  koff = (lane >> 4) * 2;
  const float* p  = X + (size_t)row * DIMD + koff;

  v8f c = {};
  // 512 WMMAs, unrolled x4: 4 global_load_b64 per lane per iteration.
  for (int k = 0; k < DIMD; k += 16) {
    v2f a0 = *(const v2f*)(p + k + 0);
    v2f a1 = *(const v2f*)(p + k + 4);
    v2f a2 = *(const v2f*)(p + k + 8);
    v2f a3 = *(const v2f*)(p + k + 12);
    c = wmma_f32(a0, a0, c);
    c = wmma_f32(a1, a1, c);
    c = wmma_f32(a2, a2, c);
    c = wmma_f32(a3, a3, c);
  }

  // C/D layout: VGPR g, lanes 0-15  -> G[m=g  ][n=lane]
  //             VGPR g, lanes 16-31 -> G[m=g+8][n=lane-16]
  // Diagonal G[i][i]: i<8 at (vgpr=i, lane=i); i>=8 at (vgpr=i-8, lane=i+16).
  #pragma unroll
  for (int g = 0; g < 8; ++g) {
    if (lane == g)      diag_s[wid][g]     = c[g];
    if (lane == g + 24) diag_s[wid][g + 8] = c[g];
  }
  __syncthreads();

  if (lane < 16) {
    float d = diag_s[wid][lane];
    invn_s[wid][lane] = 1.0f / fmaxf(sqrtf(d), NORM_EPS);
  }
  __syncthreads();

  const int   n     = lane & 15;
  const float inv_n = invn_s[wid][n];
  const int   nblk  = n >> 2;
  const bool  hi    = (lane >= 16);

  float acc = 0.0f;
  #pragma unroll
  for (int g = 0; g < 8; ++g) {
    const int m = (hi ? 8 : 0) + g;            // row index of c[g] for this lane
    if ((m >> 2) == nblk) {                    // same 4-row chunk only
      const float inv_m = hi ? invn_s[wid][g + 8] : invn_s[wid][g];
      float sim = c[g] * inv_m * inv_n;
      sim = fminf(fmaxf(sim, CLAMP_MIN), CLAMP_MAX);
      acc += -logf(1.0f - sim);
    }
  }

  // wave32 tree reduce
  #pragma unroll
  for (int off = 16; off >= 1; off >>= 1)
    acc += __shfl_xor(acc, off, 32);

  if (lane == 0)
    partial[gwave] = acc * (1.0f / 16.0f);     // mean over 4x4 per chunk
}

__global__ __launch_bounds__(1024)
void reduce_kernel(const float* __restrict__ partial, float* __restrict__ out,
                   int n) {
  __shared__ float s[1024];
  const int t = threadIdx.x;
  s[t] = (t < n) ? partial[t] : 0.0f;
  __syncthreads();
  #pragma unroll
  for (int off = 512; off >= 1; off >>= 1) {
    if (t < off) s[t] += s[t + off];
    __syncthreads();
  }
  if (t == 0) out[0] = s[0];
}

extern "C" void kernel_launch(void* const* d_in, const int* in_sizes, int n_in,
                              void* d_out, int out_size, void* d_ws,
                              size_t ws_size, hipStream_t stream) {
  (void)n_in; (void)out_size; (void)ws_size;
  const float* X   = (const float*)d_in[0];
  float*       out = (float*)d_out;
  float*       prt = (float*)d_ws;

  const int B      = in_sizes[0] / DIMD;          // 16384 rows
  const int nwaves = B / ROWS_PER_WAVE;           // 1024
  const int blocks = nwaves / WAVES_PER_BLOCK;    // 128

  gram_loss_kernel<<<blocks, BLOCK_THREADS, 0, stream>>>(X, prt);
  reduce_kernel<<<1, 1024, 0, stream>>>(prt, out, nwaves);
}